// STGCNN_15814069583907
// MI455X (gfx1250) — compile-verified
//
#include <hip/hip_runtime.h>
#include <hip/hip_bf16.h>

// Problem constants (match reference)
#define BB   4
#define PP   8
#define NN   4096
#define EE   65536
#define CIN  64
#define COUT 64

typedef __attribute__((ext_vector_type(2))) float v2f;
typedef __attribute__((ext_vector_type(8))) float v8f;

// ---------------------------------------------------------------------------
// 1) deg = 1.0 (self loop)
__global__ __launch_bounds__(256) void deg_init_k(float* __restrict__ deg) {
    int i = blockIdx.x * blockDim.x + threadIdx.x;   // over BB*NN
    if (i < BB * NN) deg[i] = 1.0f;
}

// 2) deg[dst] += 1 per edge
__global__ __launch_bounds__(256) void deg_accum_k(const int* __restrict__ ei,
                                                   float* __restrict__ deg) {
    int idx = blockIdx.x * blockDim.x + threadIdx.x; // over BB*EE
    if (idx >= BB * EE) return;
    int b = idx / EE;
    int e = idx - b * EE;
    int dst = ei[(b * 2 + 1) * EE + e];
    atomicAdd(&deg[b * NN + dst], 1.0f);
}

// 3) dinv = rsqrt(deg)   (deg >= 1 always, matches where(deg>0, rsqrt, 0))
__global__ __launch_bounds__(256) void dinv_k(const float* __restrict__ deg,
                                              float* __restrict__ dinv) {
    int i = blockIdx.x * blockDim.x + threadIdx.x;
    if (i < BB * NN) dinv[i] = rsqrtf(deg[i]);
}

// ---------------------------------------------------------------------------
// 4) h[b][n][p][d] = sum_c x[b][p][n][c] * W[c][d], via V_WMMA_F32_16X16X4_F32.
//    GEMM rows r = n*PP + p (so h rows are contiguous [NN*PP, COUT]).
//    One wave computes one 16x16 tile, chaining 16 K=4 WMMAs (exact fp32).
//
//    ISA 32-bit operand layouts (wave32):
//      A 16x4 : lane m=lane&15; VGPR0 = A[m][k0 + 2*(lane>>4)], VGPR1 = +1
//      B 4x16 : lane n=lane&15; VGPR0 = B[k0 + 2*(lane>>4)][n], VGPR1 = +1
//      C/D    : VGPR v, lane L -> M = v + 8*(L>>4), N = L&15
__global__ __launch_bounds__(256) void gemm_wmma_k(const float* __restrict__ x,
                                                   const float* __restrict__ W,
                                                   float* __restrict__ h) {
    const int ROWT = (NN * PP) / 16;  // 2048 row tiles per batch
    const int COLT = COUT / 16;       // 4   col tiles

    int wave = (blockIdx.x * blockDim.x + threadIdx.x) >> 5;
    int lane = threadIdx.x & 31;
    int b = wave / (ROWT * COLT);
    int t = wave - b * (ROWT * COLT);
    int rowTile = t / COLT;
    int colTile = t - rowTile * COLT;

    int m    = lane & 15;
    int half = lane >> 4;                 // 0: lanes 0-15, 1: lanes 16-31

    int row = rowTile * 16 + m;           // r in [0, NN*PP)
    int n   = row / PP;
    int p   = row - n * PP;
    const float* __restrict__ xrow =
        x + (((size_t)b * PP + p) * NN + n) * CIN;

    int col = colTile * 16 + (lane & 15);

    v8f acc = {};
    #pragma unroll
    for (int k0 = 0; k0 < CIN; k0 += 4) {
        int ka = k0 + 2 * half;
        v2f av; av.x = xrow[ka];
                av.y = xrow[ka + 1];
        v2f bv; bv.x = W[ka * COUT + col];
                bv.y = W[(ka + 1) * COUT + col];
        acc = __builtin_amdgcn_wmma_f32_16x16x4_f32(
            /*neg_a=*/false, av, /*neg_b=*/false, bv,
            /*c_mod=*/(short)0, acc, /*reuse_a=*/false, /*reuse_b=*/false);
    }

    size_t baseRow = (size_t)b * (NN * PP) + rowTile * 16;
    int d = colTile * 16 + (lane & 15);
    #pragma unroll
    for (int v = 0; v < 8; ++v) {
        size_t rr = baseRow + v + 8 * half;
        h[rr * COUT + d] = acc[v];
    }
}

// ---------------------------------------------------------------------------
// 5) out[b][p][n][c] = dinv[n]^2 * h[b][n][p][c]   (self-loop term; fully
//    initializes d_out each call)
__global__ __launch_bounds__(256) void selfloop_init_k(const float* __restrict__ h,
                                                       const float* __restrict__ dinv,
                                                       float* __restrict__ out) {
    size_t idx = (size_t)blockIdx.x * blockDim.x + threadIdx.x; // BB*PP*NN*COUT
    int c = (int)(idx & (COUT - 1));
    size_t t = idx >> 6;
    int n = (int)(t % NN); t /= NN;
    int p = (int)(t % PP);
    int b = (int)(t / PP);
    float dv = dinv[b * NN + n];
    out[idx] = dv * dv * h[(((size_t)b * NN + n) * PP + p) * COUT + c];
}

// 6) per edge: out[b][p][dst][c] += dinv[src]*dinv[dst] * h[b][src][p][c]
//    One wave per edge: gathers the contiguous 512-float h[src] row
//    (coalesced B128-class loads from L2) and issues contiguous
//    global_atomic_add_f32 bursts per p-slice.
__global__ __launch_bounds__(256) void edge_scatter_k(const int* __restrict__ ei,
                                                      const float* __restrict__ h,
                                                      const float* __restrict__ dinv,
                                                      float* __restrict__ out) {
    int wave = (blockIdx.x * blockDim.x + threadIdx.x) >> 5;
    int lane = threadIdx.x & 31;
    if (wave >= BB * EE) return;
    int b = wave / EE;
    int e = wave - b * EE;
    int src = ei[(b * 2 + 0) * EE + e];
    int dst = ei[(b * 2 + 1) * EE + e];
    float w = dinv[b * NN + src] * dinv[b * NN + dst];

    const float* __restrict__ hs = h + (((size_t)b * NN + src) * PP) * COUT;
    #pragma unroll
    for (int k = 0; k < (PP * COUT) / 32; ++k) {   // 16 iterations
        int f = lane + 32 * k;                     // 0..511
        int p = f >> 6;                            // frame
        int c = f & (COUT - 1);                    // channel
        atomicAdd(&out[(((size_t)(b * PP + p)) * NN + dst) * COUT + c],
                  w * hs[f]);
    }
}

// 7) out = PReLU(out + bias)
__global__ __launch_bounds__(256) void bias_prelu_k(float* __restrict__ out,
                                                    const float* __restrict__ bias,
                                                    const float* __restrict__ alpha) {
    size_t idx = (size_t)blockIdx.x * blockDim.x + threadIdx.x;
    float v = out[idx] + bias[idx & (COUT - 1)];
    float a = alpha[0];
    out[idx] = (v >= 0.0f) ? v : a * v;
}

// ---------------------------------------------------------------------------
extern "C" void kernel_launch(void* const* d_in, const int* in_sizes, int n_in,
                              void* d_out, int out_size, void* d_ws, size_t ws_size,
                              hipStream_t stream) {
    const float* x     = (const float*)d_in[0];   // [B,P,N,CIN]
    const int*   ei    = (const int*)d_in[1];     // [B,2,E]
    const float* W     = (const float*)d_in[2];   // [CIN,COUT]
    const float* bias  = (const float*)d_in[3];   // [COUT]
    const float* alpha = (const float*)d_in[4];   // [1]
    float* out = (float*)d_out;                   // [B,P,N,COUT]

    // Workspace layout
    char* ws = (char*)d_ws;
    float* h    = (float*)ws;                                     // 32 MB
    float* deg  = (float*)(ws + (size_t)BB * NN * PP * COUT * 4); // 64 KB
    float* dinv = deg + BB * NN;                                  // 64 KB
    (void)ws_size; (void)in_sizes; (void)n_in; (void)out_size;

    // 1) deg = 1
    deg_init_k<<<(BB * NN + 255) / 256, 256, 0, stream>>>(deg);
    // 2) deg[dst] += 1 per edge
    deg_accum_k<<<(BB * EE + 255) / 256, 256, 0, stream>>>(ei, deg);
    // 3) dinv = rsqrt(deg)
    dinv_k<<<(BB * NN + 255) / 256, 256, 0, stream>>>(deg, dinv);
    // 4) h = x @ W via WMMA  (BB * 2048 * 4 waves, 8 waves per 256-thr block)
    {
        int waves = BB * ((NN * PP) / 16) * (COUT / 16);
        gemm_wmma_k<<<waves / 8, 256, 0, stream>>>(x, W, h);
    }
    // 5) out = dinv^2 * h  (self loops; full d_out init)
    {
        size_t total = (size_t)BB * PP * NN * COUT;
        selfloop_init_k<<<(unsigned)(total / 256), 256, 0, stream>>>(h, dinv, out);
    }
    // 6) edge scatter with float atomics (one wave per edge)
    {
        int waves = BB * EE;
        edge_scatter_k<<<waves / 8, 256, 0, stream>>>(ei, h, dinv, out);
    }
    // 7) bias + PReLU in place
    {
        size_t total = (size_t)BB * PP * NN * COUT;
        bias_prelu_k<<<(unsigned)(total / 256), 256, 0, stream>>>(out, bias, alpha);
    }
}